// Encoder_86629490361016
// MI455X (gfx1250) — compile-verified
//
#include <hip/hip_runtime.h>
#include <hip/hip_bf16.h>
#include <math.h>

typedef float v2f __attribute__((ext_vector_type(2)));
typedef float v8f __attribute__((ext_vector_type(8)));

#define KPAD 176          // 165 padded to multiple of 16
#define NROWS 32768       // B * F_SO3 = 2048*16
#define NG 4000
#define NE 4608
#define SO3 165
#define BATCH 2048

__device__ __forceinline__ v8f wmma4(v2f a, v2f b, v8f c) {
  // D = A(16x4,f32) * B(4x16,f32) + C(16x16,f32)
  return __builtin_amdgcn_wmma_f32_16x16x4_f32(false, a, false, b, (short)0, c,
                                               false, false);
}

// ---------------------------------------------------------------- psi1/psi2
__global__ void psi_kernel(const float* __restrict__ Y, const float* __restrict__ w_s2,
                           const float* __restrict__ Dk, const float* __restrict__ w_so3,
                           float* __restrict__ psi1, float* __restrict__ psi2) {
  int t = blockIdx.x * blockDim.x + threadIdx.x;
  if (t < 16 * 25) {
    int y = t / 25, i = t - y * 25;
    float acc = 0.f;
    for (int n = 0; n < 48; ++n) acc += Y[n * 25 + i] * w_s2[y * 48 + n];
    psi1[t] = acc * 0.14433756729740643f;  // 1/sqrt(48)
  }
  int t2 = t - 16 * 25;
  if (t2 >= 0 && t2 < 16 * 165) {
    int xc = t2 / 165, i = t2 - xc * 165;
    float acc = 0.f;
    for (int n = 0; n < 192; ++n) acc += Dk[n * 165 + i] * w_so3[xc * 192 + n];
    psi2[t2] = acc * 0.07216878364870323f;  // 1/sqrt(192)
  }
}

// ------------------------------------------------- h (padded, 32768 x 176)
__global__ void build_hp_kernel(const float* __restrict__ x,
                                const float* __restrict__ psi1,
                                float* __restrict__ hp) {
  int t = blockIdx.x * 256 + threadIdx.x;
  if (t >= NROWS * KPAD) return;
  int bf = t / KPAD, i = t - bf * KPAD;
  int b = bf >> 4, y = bf & 15;
  float val = 0.f;
  if (i < SO3) {
    int s, q, d;
    if (i < 1)       { s = 0;  q = 0;  d = 1; }
    else if (i < 10) { s = 1;  q = 1;  d = 3; }
    else if (i < 35) { s = 4;  q = 10; d = 5; }
    else if (i < 84) { s = 9;  q = 35; d = 7; }
    else             { s = 16; q = 84; d = 9; }
    int local = i - q;
    int j = local / d, m = local - j * d;
    val = x[b * 25 + s + m] * psi1[y * 25 + s + j];
  }
  hp[t] = val;
}

// ------------------------------------------------- pad D_act -> (4000,176)
__global__ void pad_dact_kernel(const float* __restrict__ D_act, float* __restrict__ Dp) {
  int t = blockIdx.x * 256 + threadIdx.x;
  if (t >= NG * KPAD) return;
  int g = t / KPAD, i = t - g * KPAD;
  Dp[t] = (i < SO3) ? D_act[g * SO3 + i] : 0.f;
}

// -------------------------------------------- pad eval_wigners -> (176,4608)
__global__ void pad_ev_kernel(const float* __restrict__ ev, float* __restrict__ evP) {
  int t = blockIdx.x * 256 + threadIdx.x;
  if (t >= KPAD * NE) return;
  int k = t / NE, e = t - k * NE;
  evP[t] = (k < SO3) ? ev[k * NE + e] : 0.f;
}

// ------------------------------------------------------------------------
// Fused: h2 = (relu(hp @ Dp^T / sqrt(165))) @ Dp * sqrt(165)/4000
// 512 blocks x 128 threads (4 waves). Each block: 64 rows, full 176 cols.
// A operand (the wave's 16 rows of hp) lives in 88 VGPRs for the whole
// g-loop; B operands are software-pipelined out of LDS (explicit double
// buffering so DS loads stay ahead of the consuming WMMAs).
// ------------------------------------------------------------------------
__global__ void __launch_bounds__(128) fused_act_kernel(const float* __restrict__ hp,
                                                        const float* __restrict__ Dp,
                                                        float* __restrict__ h2p) {
  __shared__ float dB[16 * 177];      // D_act chunk rows (16 x 176)
  __shared__ float gbuf[4 * 16 * 17]; // wave-private relu(g) 16x16
  const int tid = threadIdx.x;
  const int w = tid >> 5;
  const int lane = tid & 31;
  const int half = lane >> 4;   // 0: K{0,1}, 1: K{2,3}
  const int idx = lane & 15;    // M for A, N for B/C
  const int rowBase = blockIdx.x * 64;

  // hoist the wave's A fragments for the entire K=176 loop into registers
  v2f afrag[44];
  {
    const float* ar = hp + (size_t)(rowBase + w * 16 + idx) * KPAD + 2 * half;
#pragma unroll
    for (int kk = 0; kk < 44; ++kk) {
      afrag[kk].x = ar[4 * kk];
      afrag[kk].y = ar[4 * kk + 1];
    }
  }

  v8f acc[11] = {};
  float* gw = gbuf + w * 16 * 17;
  const float* brow = dB + idx * 177;

  for (int gbase = 0; gbase < NG; gbase += 16) {
    __syncthreads();  // protect dB against stage-2 readers of previous chunk
    for (int t = tid; t < 16 * KPAD; t += 128) {
      int r = t / KPAD, c = t - r * KPAD;
      dB[r * 177 + c] = Dp[(size_t)(gbase + r) * KPAD + c];
    }
    __syncthreads();

    // ---- stage 1: g16 = hp_rows(16x176) @ Dp_chunk^T(176x16)
    // groups of 4 k-steps, double-buffered B fragments
    v8f c1 = {};
    {
      v2f n0, n1, n2, n3;
      {
        const int k0 = 2 * half;
        n0.x = brow[k0];      n0.y = brow[k0 + 1];
        n1.x = brow[k0 + 4];  n1.y = brow[k0 + 5];
        n2.x = brow[k0 + 8];  n2.y = brow[k0 + 9];
        n3.x = brow[k0 + 12]; n3.y = brow[k0 + 13];
      }
#pragma unroll
      for (int grp = 0; grp < 11; ++grp) {
        v2f c0 = n0, cA = n1, cB = n2, cC = n3;
        if (grp < 10) {
          const int k0 = (grp + 1) * 16 + 2 * half;
          n0.x = brow[k0];      n0.y = brow[k0 + 1];
          n1.x = brow[k0 + 4];  n1.y = brow[k0 + 5];
          n2.x = brow[k0 + 8];  n2.y = brow[k0 + 9];
          n3.x = brow[k0 + 12]; n3.y = brow[k0 + 13];
        }
        c1 = wmma4(afrag[grp * 4 + 0], c0, c1);
        c1 = wmma4(afrag[grp * 4 + 1], cA, c1);
        c1 = wmma4(afrag[grp * 4 + 2], cB, c1);
        c1 = wmma4(afrag[grp * 4 + 3], cC, c1);
      }
    }

    // relu + scale, re-layout via LDS (C-layout -> A-layout)
#pragma unroll
    for (int r = 0; r < 8; ++r) {
      float v = c1[r] * 0.07784989441615351f;  // 1/sqrt(165)
      gw[(r + 8 * half) * 17 + idx] = v > 0.f ? v : 0.f;
    }

    // ---- stage 2: acc += g16(16x16) @ Dp_chunk(16x176)
    // prefetch all 4 A fragments, 2-deep pipeline on B fragments
    {
      v2f a2[4];
#pragma unroll
      for (int kk = 0; kk < 4; ++kk) {
        const int k0 = 4 * kk + 2 * half;
        a2[kk].x = gw[idx * 17 + k0];
        a2[kk].y = gw[idx * 17 + k0 + 1];
      }
      v2f nb, nb2;
      {
        const int k0 = 2 * half;
        nb.x  = dB[k0 * 177 + idx];
        nb.y  = dB[(k0 + 1) * 177 + idx];
        nb2.x = dB[k0 * 177 + 16 + idx];
        nb2.y = dB[(k0 + 1) * 177 + 16 + idx];
      }
#pragma unroll
      for (int j = 0; j < 44; ++j) {
        const int kk = j / 11, t = j - kk * 11;
        v2f cb = nb;
        nb = nb2;
        if (j < 42) {
          const int j2 = j + 2;
          const int kk2 = j2 / 11, t2 = j2 - kk2 * 11;
          const int k0 = 4 * kk2 + 2 * half;
          nb2.x = dB[k0 * 177 + t2 * 16 + idx];
          nb2.y = dB[(k0 + 1) * 177 + t2 * 16 + idx];
        }
        acc[t] = wmma4(a2[kk], cb, acc[t]);
      }
    }
  }

  const float sc = 0.003211308144666282f;  // sqrt(165)/4000
#pragma unroll
  for (int t = 0; t < 11; ++t)
#pragma unroll
    for (int r = 0; r < 8; ++r) {
      int grow = rowBase + w * 16 + r + 8 * half;
      h2p[(size_t)grow * KPAD + t * 16 + idx] = acc[t][r] * sc;
    }
}

// --------------------------------------------- so3->so3 linear + harmonics
__global__ void so3lin_kernel(const float* __restrict__ h2p, const float* __restrict__ psi2,
                              float* __restrict__ harmP, float* __restrict__ harm_out) {
  int t = blockIdx.x * 256 + threadIdx.x;
  if (t >= BATCH * KPAD) return;
  int b = t / KPAD, i = t - b * KPAD;
  if (i >= SO3) { harmP[t] = 0.f; return; }
  int q, d;
  if (i < 1)       { q = 0;  d = 1; }
  else if (i < 10) { q = 1;  d = 3; }
  else if (i < 35) { q = 10; d = 5; }
  else if (i < 84) { q = 35; d = 7; }
  else             { q = 84; d = 9; }
  int local = i - q;
  int v = local / d, m = local - v * d;
  float acc = 0.f;
  for (int xch = 0; xch < 16; ++xch) {
    const float* hrow = h2p + (size_t)(b * 16 + xch) * KPAD + q;
    const float* prow = psi2 + xch * SO3 + q;
    for (int u = 0; u < d; ++u) acc += hrow[u * d + m] * prow[u * d + v];
  }
  acc *= 1.0f / sqrtf(16.0f * (float)d);
  harmP[t] = acc;
  harm_out[b * SO3 + i] = acc;
}

// --------------------------------- logits = harmP @ evP  (WMMA, into probs)
__global__ void __launch_bounds__(128) logits_kernel(const float* __restrict__ harmP,
                                                     const float* __restrict__ evP,
                                                     float* __restrict__ probs) {
  const int tid = threadIdx.x;
  const int w = tid >> 5, lane = tid & 31;
  const int half = lane >> 4, idx = lane & 15;
  const int rowBase = blockIdx.x * 16;
  const int colBase = (blockIdx.y * 4 + w) * 16;
  v8f c = {};
  const float* arow = harmP + (size_t)(rowBase + idx) * KPAD;
#pragma unroll 4
  for (int kk = 0; kk < 44; ++kk) {
    const int k0 = 4 * kk + 2 * half;
    v2f a; a.x = arow[k0]; a.y = arow[k0 + 1];
    v2f b;
    b.x = evP[(size_t)k0 * NE + colBase + idx];
    b.y = evP[(size_t)(k0 + 1) * NE + colBase + idx];
    c = wmma4(a, b, c);
  }
#pragma unroll
  for (int r = 0; r < 8; ++r) {
    int row = rowBase + r + 8 * half;
    probs[(size_t)row * NE + colBase + idx] = c[r];
  }
}

// ------------------------------------------------------- softmax (in place)
__global__ void softmax_kernel(float* __restrict__ probs) {
  __shared__ float red[256];
  float* row = probs + (size_t)blockIdx.x * NE;
  float m = -3.0e38f;
  for (int e = threadIdx.x; e < NE; e += 256) m = fmaxf(m, row[e]);
  red[threadIdx.x] = m; __syncthreads();
  for (int off = 128; off > 0; off >>= 1) {
    if (threadIdx.x < off) red[threadIdx.x] = fmaxf(red[threadIdx.x], red[threadIdx.x + off]);
    __syncthreads();
  }
  float rowmax = red[0]; __syncthreads();
  float s = 0.f;
  for (int e = threadIdx.x; e < NE; e += 256) s += expf(row[e] - rowmax);
  red[threadIdx.x] = s; __syncthreads();
  for (int off = 128; off > 0; off >>= 1) {
    if (threadIdx.x < off) red[threadIdx.x] += red[threadIdx.x + off];
    __syncthreads();
  }
  float inv = 1.0f / red[0]; __syncthreads();
  for (int e = threadIdx.x; e < NE; e += 256) row[e] = expf(row[e] - rowmax) * inv;
}

// -------------------------------------- 5th largest of probs row 0 (1 block)
__global__ void top5_kernel(const float* __restrict__ probs, float* __restrict__ fifth) {
  __shared__ float svals[256];
  __shared__ int sidx[256];
  __shared__ int ssel[5];
  for (int it = 0; it < 5; ++it) {
    float best = -3.0e38f; int bidx = -1;
    for (int e = threadIdx.x; e < NE; e += 256) {
      bool skip = false;
      for (int s = 0; s < it; ++s) if (ssel[s] == e) skip = true;
      float v = probs[e];
      if (!skip && v > best) { best = v; bidx = e; }
    }
    svals[threadIdx.x] = best; sidx[threadIdx.x] = bidx;
    __syncthreads();
    for (int off = 128; off > 0; off >>= 1) {
      if (threadIdx.x < off && svals[threadIdx.x + off] > svals[threadIdx.x]) {
        svals[threadIdx.x] = svals[threadIdx.x + off];
        sidx[threadIdx.x] = sidx[threadIdx.x + off];
      }
      __syncthreads();
    }
    if (threadIdx.x == 0) {
      ssel[it] = sidx[0];
      if (it == 4) *fifth = svals[0];
    }
    __syncthreads();
  }
}

__global__ void threshold_kernel(float* __restrict__ probs, const float* __restrict__ fifth) {
  size_t t = (size_t)blockIdx.x * 256 + threadIdx.x;
  if (t >= (size_t)BATCH * NE) return;
  float f = *fifth;
  float p = probs[t];
  if (p < f) probs[t] = 0.f;
}

extern "C" void kernel_launch(void* const* d_in, const int* in_sizes, int n_in,
                              void* d_out, int out_size, void* d_ws, size_t ws_size,
                              hipStream_t stream) {
  const float* x      = (const float*)d_in[0];
  const float* w_s2   = (const float*)d_in[1];
  const float* Y      = (const float*)d_in[2];
  const float* w_so3  = (const float*)d_in[3];
  const float* Dk     = (const float*)d_in[4];
  const float* D_act  = (const float*)d_in[5];
  const float* evw    = (const float*)d_in[6];

  float* out      = (float*)d_out;
  float* harm_out = out;                       // 2048*165
  float* probs    = out + (size_t)BATCH * SO3; // 2048*4608

  float* ws    = (float*)d_ws;
  float* psi1  = ws;                  // 400
  float* psi2  = psi1 + 400;          // 2640
  float* hp    = psi2 + 2640;         // 32768*176
  float* Dp    = hp + (size_t)NROWS * KPAD;   // 4000*176
  float* h2p   = Dp + (size_t)NG * KPAD;      // 32768*176
  float* harmP = h2p + (size_t)NROWS * KPAD;  // 2048*176
  float* evP   = harmP + (size_t)BATCH * KPAD;// 176*4608
  float* fifth = evP + (size_t)KPAD * NE;     // 1

  psi_kernel<<<12, 256, 0, stream>>>(Y, w_s2, Dk, w_so3, psi1, psi2);
  build_hp_kernel<<<(NROWS * KPAD + 255) / 256, 256, 0, stream>>>(x, psi1, hp);
  pad_dact_kernel<<<(NG * KPAD + 255) / 256, 256, 0, stream>>>(D_act, Dp);
  pad_ev_kernel<<<(KPAD * NE + 255) / 256, 256, 0, stream>>>(evw, evP);
  fused_act_kernel<<<NROWS / 64, 128, 0, stream>>>(hp, Dp, h2p);
  so3lin_kernel<<<(BATCH * KPAD + 255) / 256, 256, 0, stream>>>(h2p, psi2, harmP, harm_out);
  logits_kernel<<<dim3(BATCH / 16, NE / 64), 128, 0, stream>>>(harmP, evP, probs);
  softmax_kernel<<<BATCH, 256, 0, stream>>>(probs);
  top5_kernel<<<1, 256, 0, stream>>>(probs, fifth);
  threshold_kernel<<<(int)(((size_t)BATCH * NE + 255) / 256), 256, 0, stream>>>(probs, fifth);
}